// AdaptiveSparseWindowExtractor_51496657879111
// MI455X (gfx1250) — compile-verified
//
#include <hip/hip_runtime.h>
#include <hip/hip_bf16.h>
#include <math.h>
#include <stdint.h>

// Problem constants (match reference)
#define BATCH 8
#define CCH   256
#define HH    128
#define WW    128
#define HWSZ  (HH * WW)          // 16384
#define KTOP  1280               // min(int(0.1*16384), 1280)
#define WIN   5
#define PADR  2
#define PP    25                 // WIN*WIN
#define TEMP_INV 2.0f            // 1/0.5
#define LAMW  2.5f               // LAMBDA * WIN

typedef __attribute__((ext_vector_type(2)))  float v2f;
typedef __attribute__((ext_vector_type(8)))  float v8f;
typedef __attribute__((ext_vector_type(4)))  unsigned int u32x4;
typedef __attribute__((ext_vector_type(8)))  int i32x8;
typedef __attribute__((ext_vector_type(4)))  int i32x4;

// ---------------------------------------------------------------------------
// Kernel 1: per-batch softmax of (saliency + mask_logits)/TEMP -> calibrated
// Also emits the constant offsets output (50 floats) from block 0.
// ---------------------------------------------------------------------------
__global__ void k1_softmax(const float* __restrict__ sal,
                           const float* __restrict__ maskl,
                           float* __restrict__ cal,
                           float* __restrict__ offs_out) {
    __shared__ float red[256];
    const int b = blockIdx.x;
    const int tid = threadIdx.x;
    const float* sb = sal + (size_t)b * HWSZ;

    // pass 1: max of t_i = (sal+mask)*2
    float mx = -INFINITY;
    for (int i = tid; i < HWSZ; i += 256)
        mx = fmaxf(mx, (sb[i] + maskl[i]) * TEMP_INV);
    red[tid] = mx; __syncthreads();
    for (int s = 128; s > 0; s >>= 1) {
        if (tid < s) red[tid] = fmaxf(red[tid], red[tid + s]);
        __syncthreads();
    }
    mx = red[0]; __syncthreads();

    // pass 2: sum of exp(t - mx)
    float sum = 0.0f;
    for (int i = tid; i < HWSZ; i += 256)
        sum += expf((sb[i] + maskl[i]) * TEMP_INV - mx);
    red[tid] = sum; __syncthreads();
    for (int s = 128; s > 0; s >>= 1) {
        if (tid < s) red[tid] += red[tid + s];
        __syncthreads();
    }
    sum = red[0];
    const float inv = 1.0f / sum;

    // pass 3: write softmax
    for (int i = tid; i < HWSZ; i += 256)
        cal[(size_t)b * HWSZ + i] = expf((sb[i] + maskl[i]) * TEMP_INV - mx) * inv;

    // constant offsets output [1,1,P,2]: (dy, dx) per p
    if (b == 0 && tid < 2 * PP) {
        int p = tid >> 1, comp = tid & 1;
        int v = comp ? (p % WIN - PADR) : (p / WIN - PADR);
        offs_out[tid] = (float)v;
    }
}

// ---------------------------------------------------------------------------
// Kernel 2: exact sorted top-K via rank counting (matches jax.lax.top_k:
// descending value, ties broken by smaller index first). One block per batch,
// scores cached in LDS (uniform j => bank broadcast).
// ---------------------------------------------------------------------------
__global__ void k2_topk(const float* __restrict__ cal, int* __restrict__ topk) {
    __shared__ float v[HWSZ];            // 64 KB
    const int b = blockIdx.x;
    const int tid = threadIdx.x;         // 1024 threads
    for (int i = tid; i < HWSZ; i += 1024)
        v[i] = cal[(size_t)b * HWSZ + i];
    __syncthreads();

    for (int i = tid; i < HWSZ; i += 1024) {
        const float vi = v[i];
        int rank = 0;
        for (int j = 0; j < HWSZ; ++j) {
            const float vj = v[j];
            rank += (vj > vi) || ((vj == vi) && (j < i));
        }
        if (rank < KTOP) topk[(size_t)b * KTOP + rank] = i;
    }
}

// ---------------------------------------------------------------------------
// Kernel 3: per-window weights.  One wave (32 threads) handles 16 windows.
// mean over P=25 scores computed with V_WMMA_F32_16X16X4_F32 chained 7x:
//   D = A(16x4, all ones) x B(4x16, scores per window column) + C
// => element 0 of the accumulator in every lane = sum of its window's scores.
// Then w[b,k,p] = sigmoid(gamma*(s - mean)) * exp(-dist(p)/2.5).
// Also writes topk_coords (as float) output.
// ---------------------------------------------------------------------------
__global__ void k3_weights(const float* __restrict__ cal,
                           const int* __restrict__ topk,
                           const float* __restrict__ gamma_p,
                           float* __restrict__ wgt,
                           float* __restrict__ coords) {
    const int lane = threadIdx.x;        // 0..31, blockDim.x == 32
    const int w    = lane & 15;          // window slot within the wave
    const int half = lane >> 4;
    const int bk0  = blockIdx.x * 16;    // 16 windows per block, same batch (16 | K)
    const int b    = bk0 / KTOP;
    const int bk   = bk0 + w;

    int idx = topk[bk];
    int cy = min(max(idx / WW, PADR), HH - 1 - PADR);
    int cx = min(max(idx % WW, PADR), WW - 1 - PADR);

    if (lane < 16) {
        coords[(size_t)bk * 2 + 0] = (float)cy;
        coords[(size_t)bk * 2 + 1] = (float)cx;
    }

    const float* calb = cal + (size_t)b * HWSZ;

    // B-matrix layout (32-bit, 4x16): VGPR0 lanes0-15 -> K=0, lanes16-31 -> K=2;
    // VGPR1 -> K=1 / K=3.  So this lane supplies p = 4*ck + 2*half (+1).
    v2f a; a[0] = 1.0f; a[1] = 1.0f;     // A = all ones (uniform across lanes)
    v8f acc = {};
    for (int ck = 0; ck < 7; ++ck) {     // covers p = 0..27 (zero-pad p >= 25)
        const int p0 = 4 * ck + 2 * half;
        const int p1 = p0 + 1;
        v2f bb;
        bb[0] = (p0 < PP) ? calb[(cy + p0 / WIN - PADR) * WW + (cx + p0 % WIN - PADR)] : 0.0f;
        bb[1] = (p1 < PP) ? calb[(cy + p1 / WIN - PADR) * WW + (cx + p1 % WIN - PADR)] : 0.0f;
        acc = __builtin_amdgcn_wmma_f32_16x16x4_f32(
                  false, a, false, bb, (short)0, acc, false, false);
    }
    const float mean = acc[0] * (1.0f / 25.0f);
    const float gm = gamma_p[0];

    // split the 25 points across the two half-waves (each lane owns window w)
    const int pstart = half ? 13 : 0;
    const int pend   = half ? PP : 13;
    for (int p = pstart; p < pend; ++p) {
        const int dy = p / WIN - PADR;
        const int dx = p % WIN - PADR;
        const float s = calb[(cy + dy) * WW + (cx + dx)];
        const float sig = 1.0f / (1.0f + expf(-gm * (s - mean)));
        const float dw  = expf(-sqrtf((float)(dy * dy + dx * dx)) / LAMW);
        wgt[(size_t)bk * PP + p] = sig * dw;
    }
}

// ---------------------------------------------------------------------------
// Kernel 4: main bandwidth kernel. One block (256 thr) per window (b,k).
// Wave 0 issues a single TDM tensor_load_to_lds of the 3D tile
// [c=256][wy=5][wx=5] (strides 16384/128/1 elements) into LDS, waits on
// TENSORcnt, then all threads scale by w[b,k,p] and store C-contiguous.
// LDS reads at stride 25 floats (gcd(25,64)=1) are bank-conflict free;
// global stores are fully coalesced.
// ---------------------------------------------------------------------------
__global__ void k4_gather(const float* __restrict__ feat,
                          const int* __restrict__ topk,
                          const float* __restrict__ wgt,
                          float* __restrict__ patches) {
    __shared__ float tile[CCH * PP];     // 25.6 KB, layout [c][wy][wx]
    __shared__ float wW[PP];
    const int bk  = blockIdx.x;
    const int b   = bk / KTOP;
    const int tid = threadIdx.x;         // 256 threads

    if (tid < PP) wW[tid] = wgt[(size_t)bk * PP + tid];

    if (tid < 32) {                      // wave 0 issues the DMA
        const int idx = __builtin_amdgcn_readfirstlane(topk[bk]);
        const int cy = min(max(idx / WW, PADR), HH - 1 - PADR);
        const int cx = min(max(idx % WW, PADR), WW - 1 - PADR);
        const uint64_t gaddr = (uint64_t)(uintptr_t)(
            feat + (size_t)b * CCH * HWSZ + (size_t)(cy - PADR) * WW + (cx - PADR));
        const uint32_t ldsa = (uint32_t)(uintptr_t)(&tile[0]);

        u32x4 g0;                        // D# group 0
        g0[0] = 1u;                                              // count=1, user mode
        g0[1] = ldsa;                                            // lds_addr
        g0[2] = (uint32_t)gaddr;                                 // global_addr[31:0]
        g0[3] = (uint32_t)((gaddr >> 32) & 0x1FFFFFFu)           // global_addr[56:32]
                | 0x80000000u;                                   // type=2 ("image")
        i32x8 g1;                        // D# group 1
        g1[0] = (int)(2u << 16);                                 // data_size=4B
        g1[1] = (int)(((uint32_t)WW & 0xFFFFu) << 16);           // tensor_dim0 lo16
        g1[2] = (int)(((uint32_t)WW >> 16) |
                      (((uint32_t)HH & 0xFFFFu) << 16));         // dim0 hi | dim1 lo
        g1[3] = (int)(((uint32_t)HH >> 16) | (5u << 16));        // dim1 hi | tile_dim0=5
        g1[4] = (int)(5u | (256u << 16));                        // tile_dim1=5 | tile_dim2=256
        g1[5] = (int)WW;                                         // tensor_dim0_stride = 128
        g1[6] = (int)(((uint32_t)HWSZ & 0xFFFFu) << 16);         // dim1_stride lo16 = 16384
        g1[7] = (int)((uint32_t)HWSZ >> 16);                     // dim1_stride hi
        i32x4 g2;                        // D# group 2 (3D tensor)
        g2[0] = CCH;                                             // tensor_dim2 = 256
        g2[1] = 0; g2[2] = 0; g2[3] = 0;
        i32x4 g3; g3[0] = 0; g3[1] = 0; g3[2] = 0; g3[3] = 0;    // group 3 unused
        i32x8 gx;                        // extra operand in 6-arg builtin (zero fill)
        gx[0] = 0; gx[1] = 0; gx[2] = 0; gx[3] = 0;
        gx[4] = 0; gx[5] = 0; gx[6] = 0; gx[7] = 0;

        __builtin_amdgcn_tensor_load_to_lds(g0, g1, g2, g3, gx, 0);
        __builtin_amdgcn_s_wait_tensorcnt(0);
    }
    __syncthreads();

    float* __restrict__ outp = patches + (size_t)bk * PP * CCH;
    const int c = tid;
    #pragma unroll
    for (int p = 0; p < PP; ++p) {
        const float v = tile[c * PP + p];
        outp[p * CCH + c] = v * wW[p];
    }
}

// ---------------------------------------------------------------------------
// Host launcher
// ---------------------------------------------------------------------------
extern "C" void kernel_launch(void* const* d_in, const int* in_sizes, int n_in,
                              void* d_out, int out_size, void* d_ws, size_t ws_size,
                              hipStream_t stream) {
    const float* feat  = (const float*)d_in[0];   // [B,C,H,W]
    const float* sal   = (const float*)d_in[1];   // [B,H,W]
    const float* maskl = (const float*)d_in[2];   // [1,H,W]
    const float* gamma = (const float*)d_in[3];   // [1]

    float* out = (float*)d_out;
    float* patches = out;                                   // B*K*P*C
    float* coords  = out + (size_t)BATCH * KTOP * PP * CCH; // B*K*2
    float* offs    = coords + (size_t)BATCH * KTOP * 2;     // 50
    float* cal     = offs + 2 * PP;                         // B*H*W

    int*   topk = (int*)d_ws;                                          // B*K ints
    float* wgt  = (float*)((char*)d_ws + (size_t)BATCH * KTOP * 4);    // B*K*P floats

    k1_softmax<<<BATCH, 256, 0, stream>>>(sal, maskl, cal, offs);
    k2_topk   <<<BATCH, 1024, 0, stream>>>(cal, topk);
    k3_weights<<<(BATCH * KTOP) / 16, 32, 0, stream>>>(cal, topk, gamma, wgt, coords);
    k4_gather <<<BATCH * KTOP, 256, 0, stream>>>(feat, topk, wgt, patches);
}